// NeuralGas_34127810134273
// MI455X (gfx1250) — compile-verified
//
#include <hip/hip_runtime.h>
#include <hip/hip_bf16.h>
#include <stdint.h>

#define B_ROWS   16384
#define N_UNITS  4096
#define IN_F     128
#define TOPK     10

typedef __attribute__((ext_vector_type(2))) float v2f;
typedef __attribute__((ext_vector_type(8))) float v8f;
typedef unsigned long long u64;

// ---------------------------------------------------------------------------
// Kernel 1: squared row norms. One wave32 per row of 128 floats.
// lane loads float4 at col=lane*4, squares, wave-reduces via shfl_xor.
// ---------------------------------------------------------------------------
__global__ __launch_bounds__(256)
void rownorm_kernel(const float* __restrict__ src, float* __restrict__ out, int nrows) {
    int wave = blockIdx.x * (blockDim.x >> 5) + (threadIdx.x >> 5);
    int lane = threadIdx.x & 31;
    if (wave >= nrows) return;
    const float4* p = (const float4*)(src + (size_t)wave * IN_F);
    float4 v = p[lane];
    float s = v.x * v.x + v.y * v.y + v.z * v.z + v.w * v.w;
#pragma unroll
    for (int off = 16; off > 0; off >>= 1)
        s += __shfl_xor(s, off, 32);
    if (lane == 0) out[wave] = s;
}

// ---------------------------------------------------------------------------
// Kernel 2: distance matrix via V_WMMA_F32_16X16X4_F32.
// Block = 256 threads = 8 waves; each wave owns one 16x16 tile of d.
// Block covers 32 rows x 64 cols. x/c are L2-resident (10 MB total), so
// direct global fragment loads are the right data path; output stores are
// the real roofline term.
// ---------------------------------------------------------------------------
__global__ __launch_bounds__(256)
void dist_kernel(const float* __restrict__ x,
                 const float* __restrict__ c,
                 const float* __restrict__ x2,
                 const float* __restrict__ c2,
                 float* __restrict__ dmat) {
    const int lane  = threadIdx.x & 31;
    const int wave  = threadIdx.x >> 5;       // 0..7
    const int wm    = wave >> 2;              // 0..1
    const int wn    = wave & 3;               // 0..3
    const int m0    = blockIdx.y * 32 + wm * 16;
    const int n0    = blockIdx.x * 64 + wn * 16;
    const int l15   = lane & 15;
    const int khalf = (lane >> 4) << 1;       // 0 for lanes 0-15, 2 for 16-31

    const float* xrow = x + (size_t)(m0 + l15) * IN_F + khalf;
    const float* crow = c + (size_t)(n0 + l15) * IN_F + khalf;

    v8f acc = {};
#pragma unroll
    for (int kk = 0; kk < IN_F / 4; ++kk) {
        v2f a = *(const v2f*)(xrow + kk * 4);
        v2f b = *(const v2f*)(crow + kk * 4);
        // 8 args: (neg_a, A, neg_b, B, c_mod, C, reuse_a, reuse_b)
        acc = __builtin_amdgcn_wmma_f32_16x16x4_f32(
            false, a, false, b, (short)0, acc, false, false);
    }

    // C/D layout: VGPR r -> M = m0 + r + (lane<16 ? 0 : 8), N = n0 + (lane&15)
    const float cn = c2[n0 + l15];
    const int mhi  = (lane >> 4) << 3;
#pragma unroll
    for (int r = 0; r < 8; ++r) {
        int m = m0 + r + mhi;
        float d2 = x2[m] - 2.0f * acc[r] + cn;
        dmat[(size_t)m * N_UNITS + (n0 + l15)] = sqrtf(fmaxf(d2, 0.0f));
    }
}

// ---------------------------------------------------------------------------
// Kernel 3: per-row full argsort of 4096 distances + top-10 one-hot.
// Keys = (distBits<<32)|idx: d >= 0 so IEEE bits sort monotonically, and the
// idx low bits reproduce jnp.argsort's stable tie-break.
//
// Register-blocked bitonic sort: thread t owns keys[16t..16t+15].
//  - phases k=2..16 (all j<=8) run entirely in registers (1 LDS pass)
//  - phases k>=32: j>=16 stages in LDS (36 total), j=8..1 in registers
// => ~46 LDS passes instead of 78, and barrier count drops to match.
// 32 KB LDS/row, well inside 320 KB/WGP.
// ---------------------------------------------------------------------------
__global__ __launch_bounds__(256)
void argsort_kernel(const float* __restrict__ dmat,
                    int*   __restrict__ kout,
                    float* __restrict__ zout) {
    __shared__ u64 keys[N_UNITS];
    const int    tid  = threadIdx.x;
    const size_t base = (size_t)blockIdx.x * N_UNITS;

    // Build keys with coalesced b128 loads of the distance row.
    const float4* drow4 = (const float4*)(dmat + base);
    for (int i = tid; i < N_UNITS / 4; i += 256) {
        float4 v = drow4[i];
        int gi = 4 * i;
        keys[gi + 0] = ((u64)__float_as_uint(v.x) << 32) | (unsigned)(gi + 0);
        keys[gi + 1] = ((u64)__float_as_uint(v.y) << 32) | (unsigned)(gi + 1);
        keys[gi + 2] = ((u64)__float_as_uint(v.z) << 32) | (unsigned)(gi + 2);
        keys[gi + 3] = ((u64)__float_as_uint(v.w) << 32) | (unsigned)(gi + 3);
    }
    __syncthreads();

    const int gbase = tid * 16;

    // Phase A: k = 2,4,8,16 — every stage has j <= 8, fully in-register.
    {
        u64 e[16];
#pragma unroll
        for (int li = 0; li < 16; ++li) e[li] = keys[gbase + li];
#pragma unroll
        for (int k = 2; k <= 16; k <<= 1) {
#pragma unroll
            for (int j = k >> 1; j > 0; j >>= 1) {
#pragma unroll
                for (int li = 0; li < 16; ++li) {
                    int partner = li ^ j;
                    if (partner > li) {
                        bool up = (((gbase + li) & k) == 0);
                        u64 a = e[li], b = e[partner];
                        bool sw = up ? (a > b) : (a < b);
                        if (sw) { e[li] = b; e[partner] = a; }
                    }
                }
            }
        }
#pragma unroll
        for (int li = 0; li < 16; ++li) keys[gbase + li] = e[li];
    }
    __syncthreads();

    // Phase B: k = 32..4096.
    for (int k = 32; k <= N_UNITS; k <<= 1) {
        // Cross-block stages (j >= 16) through LDS.
        for (int j = k >> 1; j >= 16; j >>= 1) {
#pragma unroll 4
            for (int t = tid; t < N_UNITS / 2; t += 256) {
                int i   = 2 * t - (t & (j - 1));
                int ixj = i + j;
                bool up = ((i & k) == 0);
                u64 a = keys[i];
                u64 b = keys[ixj];
                bool sw = up ? (a > b) : (a < b);
                if (sw) { keys[i] = b; keys[ixj] = a; }
            }
            __syncthreads();
        }
        // Tail stages j = 8..1 inside this thread's 16-key block.
        {
            u64 e[16];
            const bool up = ((gbase & k) == 0);  // k >= 32: uniform per block
#pragma unroll
            for (int li = 0; li < 16; ++li) e[li] = keys[gbase + li];
#pragma unroll
            for (int j = 8; j > 0; j >>= 1) {
#pragma unroll
                for (int li = 0; li < 16; ++li) {
                    int partner = li ^ j;
                    if (partner > li) {
                        u64 a = e[li], b = e[partner];
                        bool sw = up ? (a > b) : (a < b);
                        if (sw) { e[li] = b; e[partner] = a; }
                    }
                }
            }
#pragma unroll
            for (int li = 0; li < 16; ++li) keys[gbase + li] = e[li];
        }
        __syncthreads();
    }

    // k output: coalesced b128 stores of the low-32 index bits.
    int4* krow4 = (int4*)(kout + base);
    for (int i = tid; i < N_UNITS / 4; i += 256) {
        int4 v;
        v.x = (int)(unsigned)keys[4 * i + 0];
        v.y = (int)(unsigned)keys[4 * i + 1];
        v.z = (int)(unsigned)keys[4 * i + 2];
        v.w = (int)(unsigned)keys[4 * i + 3];
        krow4[i] = v;
    }

    // z output: vectorized zero-fill, then 10 scattered ones.
    float4* zrow4 = (float4*)(zout + base);
    const float4 zero4 = make_float4(0.f, 0.f, 0.f, 0.f);
    for (int i = tid; i < N_UNITS / 4; i += 256) zrow4[i] = zero4;
    __syncthreads();   // zeros globally ordered before the ones
    if (tid < TOPK) {
        unsigned idx = (unsigned)keys[tid];   // low 32 bits = column index
        zout[base + idx] = 1.0f;
    }
}

// ---------------------------------------------------------------------------
extern "C" void kernel_launch(void* const* d_in, const int* in_sizes, int n_in,
                              void* d_out, int out_size, void* d_ws, size_t ws_size,
                              hipStream_t stream) {
    const float* x = (const float*)d_in[0];   // (16384, 128)
    const float* c = (const float*)d_in[1];   // (4096, 128)

    float* x2 = (float*)d_ws;                 // 16384 floats
    float* c2 = x2 + B_ROWS;                  // 4096 floats

    const size_t MN = (size_t)B_ROWS * N_UNITS;
    float* dmat = (float*)d_out;              // output 0: d
    int*   kout = (int*)((float*)d_out + MN); // output 1: k (int32 bits)
    float* zout = (float*)d_out + 2 * MN;     // output 2: z

    rownorm_kernel<<<B_ROWS / 8, 256, 0, stream>>>(x, x2, B_ROWS);
    rownorm_kernel<<<N_UNITS / 8, 256, 0, stream>>>(c, c2, N_UNITS);

    dim3 grid(N_UNITS / 64, B_ROWS / 32);     // (64, 512)
    dist_kernel<<<grid, 256, 0, stream>>>(x, c, x2, c2, dmat);

    argsort_kernel<<<B_ROWS, 256, 0, stream>>>(dmat, kout, zout);
}